// HoltWintersDecompositionLayer_11716670784226
// MI455X (gfx1250) — compile-verified
//
#include <hip/hip_runtime.h>

// Holt-Winters decomposition, B=512, T=8192, F=8, S=24 (season_length
// structurally hard-coded: state dim = S+1 = 25, padded to 32).
//
// Exact chunked affine-scan parallelization:
//   z_{t+1} = M z_t + b(p_t),  z = [level, s_{t-24..t-1}] (25-dim)
//   chunk length L=512 -> z_end = M^512 z_start + c_chunk
// Kernels: matpow (M^512), head (init level + t<24 outputs),
// pass1 (per-(b,chunk) zero-init scan -> c), combine (WMMA f32 16x16x4
// boundary GEMM Z_{k+1} = A Z_k + C_k), pass3 (exact re-scan + fused copy).

typedef __attribute__((ext_vector_type(2))) float v2f;
typedef __attribute__((ext_vector_type(8))) float v8f;

#define BATCH   512
#define TLEN    8192
#define SEAS    24
#define CHUNK   512
#define NCHUNK  16      // 15 full chunks of 512 + last chunk of 488
#define SBLOCK  256     // threads per block for pass1/pass3

// ---------------------------------------------------------------------------
// Kernel 1: A = M^512 (M padded into 32x32, zeros outside 25x25).
// Row 0:      l' = (1-a) l - a s_old            (+ a p)
// Rows 1..23: shift                              s-buffer
// Row 24:     s' = -g(1-a) l + (g a + 1 - g) s_old  (+ g(1-a) p)
// ---------------------------------------------------------------------------
__global__ __launch_bounds__(256) void matpow_kernel(const float* __restrict__ alpha,
                                                     const float* __restrict__ gamma,
                                                     float* __restrict__ Aout) {
    __shared__ float P[1024];
    __shared__ float Q[1024];
    const int tid = threadIdx.x;
    const float a = alpha[0], g = gamma[0];
    for (int idx = tid; idx < 1024; idx += 256) {
        int r = idx >> 5, c = idx & 31;
        float v = 0.0f;
        if (r == 0) {
            if (c == 0) v = 1.0f - a;
            else if (c == 1) v = -a;
        } else if (r <= 23) {
            if (c == r + 1) v = 1.0f;
        } else if (r == 24) {
            if (c == 0) v = -g * (1.0f - a);
            else if (c == 1) v = g * a + 1.0f - g;
        }
        P[idx] = v;
    }
    __syncthreads();
    for (int it = 0; it < 9; ++it) {          // M^(2^9) = M^512
        for (int idx = tid; idx < 1024; idx += 256) {
            int r = idx >> 5, c = idx & 31;
            float s = 0.0f;
#pragma unroll
            for (int j = 0; j < 32; ++j)
                s = __builtin_fmaf(P[r * 32 + j], P[j * 32 + c], s);
            Q[idx] = s;
        }
        __syncthreads();
        for (int idx = tid; idx < 1024; idx += 256) P[idx] = Q[idx];
        __syncthreads();
    }
    for (int idx = tid; idx < 1024; idx += 256) Aout[idx] = P[idx];
}

// ---------------------------------------------------------------------------
// Kernel 2: per-row init level (mean of first 24 prices) + head outputs t<24.
// ---------------------------------------------------------------------------
__global__ __launch_bounds__(32) void head_kernel(const float* __restrict__ x,
                                                  float* __restrict__ out,
                                                  float* __restrict__ lvl) {
    const int b = blockIdx.x;
    const int lane = threadIdx.x;
    const long base = (long)b * TLEN;
    float p = 0.0f;
    if (lane < SEAS) p = x[(base + lane) * 8];
    float s = p;
#pragma unroll
    for (int off = 16; off > 0; off >>= 1) s += __shfl_down(s, off, 32);
    const float il = __shfl(s, 0, 32) * (1.0f / (float)SEAS);
    if (lane == 0) lvl[b] = il;
    for (int idx = lane; idx < SEAS * 11; idx += 32) {
        int t = idx / 11, ch = idx - t * 11;
        long bt = base + t;
        float v;
        if (ch == 0)      v = x[bt * 8];            // deseason_head = price
        else if (ch <= 8) v = x[bt * 8 + ch - 1];   // copy x
        else if (ch == 9) v = il;                   // level_head
        else              v = 0.0f;                 // seasonal_head
        out[bt * 11 + ch] = v;
    }
}

// ---------------------------------------------------------------------------
// Kernel 3: pass1 — one lane per (b, chunk k<15), zero-init chunk scan -> c_k.
// Seasonal ring buffer in LDS: sbuf[slot*SBLOCK + tid] (conflict-free).
// ---------------------------------------------------------------------------
__global__ __launch_bounds__(SBLOCK) void pass1_kernel(const float* __restrict__ x,
                                                       const float* __restrict__ alpha,
                                                       const float* __restrict__ gamma,
                                                       float* __restrict__ cbuf) {
    __shared__ float sbuf[SEAS * SBLOCK];
    const int tid = threadIdx.x;
    const int gid = blockIdx.x * SBLOCK + tid;
    const int b = gid >> 4;
    const int k = gid & 15;
    if (b >= BATCH || k >= NCHUNK - 1) return;   // c_15 never consumed
    const float a = alpha[0], g = gamma[0];
    const float oma = 1.0f - a, omg = 1.0f - g;
    const int t0 = SEAS + k * CHUNK;
    const int tend = t0 + CHUNK;                 // k<15 => always full chunk
#pragma unroll
    for (int i = 0; i < SEAS; ++i) sbuf[i * SBLOCK + tid] = 0.0f;
    float l = 0.0f;
    int slot = t0 % SEAS;
    const long base = (long)b * TLEN;
    for (int t = t0; t < tend; ++t) {
        float p = x[(base + t) * 8];
        float sp = sbuf[slot * SBLOCK + tid];
        l = __builtin_fmaf(oma, l, a * (p - sp));
        float sv = __builtin_fmaf(omg, sp, g * (p - l));
        sbuf[slot * SBLOCK + tid] = sv;
        slot = (slot + 1 == SEAS) ? 0 : slot + 1;
    }
    float* c = cbuf + ((long)(b * NCHUNK + k)) * 32;
    c[0] = l;
#pragma unroll
    for (int i = 0; i < SEAS; ++i)               // s_{tend-24+i} lives at slot (tend+i)%24
        c[1 + i] = sbuf[((tend + i) % SEAS) * SBLOCK + tid];
#pragma unroll
    for (int i = SEAS + 1; i < 32; ++i) c[i] = 0.0f;
}

// ---------------------------------------------------------------------------
// Kernel 4: combine — Z_{k+1}[32x512] = A[32x32] Z_k + C_k via WMMA f32
// 16x16x4. 32 blocks x 1 wave; each block owns 16 batch columns.
// A frag (16x4): lanes 0-15 rows, VGPR0 = K{+0,+2}, VGPR1 = K{+1,+3}.
// B frag (4x16): mirrored striping (rows K{0,2}/{1,3} across lane halves).
// C/D (16x16):   VGPR j: lanes 0-15 -> M=j, lanes 16-31 -> M=j+8.
// ---------------------------------------------------------------------------
__global__ __launch_bounds__(32) void combine_kernel(const float* __restrict__ A,
                                                     const float* __restrict__ lvl,
                                                     const float* __restrict__ cbuf,
                                                     float* __restrict__ zbuf) {
    __shared__ float zt[32 * 16];                // Z_k tile, [state row][batch col]
    const int lane = threadIdx.x;
    const int lo = lane & 15;
    const int hi = lane >> 4;
    const int b0 = blockIdx.x * 16;

    // Load A fragments once (2 M-tiles x 8 K-steps).
    v2f afrag[2][8];
#pragma unroll
    for (int mb = 0; mb < 2; ++mb)
#pragma unroll
        for (int kb = 0; kb < 8; ++kb) {
            int row = mb * 16 + lo;
            int col = kb * 4 + hi * 2;
            afrag[mb][kb].x = A[row * 32 + col];
            afrag[mb][kb].y = A[row * 32 + col + 1];
        }

    // Build Z_0 = [init_level; 0...] in LDS and in global zbuf.
    for (int idx = lane; idx < 512; idx += 32) {
        int row = idx >> 4, col = idx & 15;
        float v = (row == 0) ? lvl[b0 + col] : 0.0f;
        zt[row * 16 + col] = v;
        zbuf[(long)(b0 + col) * (NCHUNK * 32) + row] = v;
    }
    __syncthreads();

    for (int k = 0; k < NCHUNK - 1; ++k) {
        // Accumulators seeded with C_k.
        v8f d0, d1;
#pragma unroll
        for (int j = 0; j < 8; ++j) {
            int m = j + hi * 8;
            long cb = ((long)(b0 + lo) * NCHUNK + k) * 32;
            d0[j] = cbuf[cb + m];
            d1[j] = cbuf[cb + 16 + m];
        }
#pragma unroll
        for (int kb = 0; kb < 8; ++kb) {
            v2f bfrag;
            int r = kb * 4 + hi * 2;
            bfrag.x = zt[r * 16 + lo];
            bfrag.y = zt[(r + 1) * 16 + lo];
            d0 = __builtin_amdgcn_wmma_f32_16x16x4_f32(false, afrag[0][kb],
                                                       false, bfrag,
                                                       (short)0, d0, false, false);
            d1 = __builtin_amdgcn_wmma_f32_16x16x4_f32(false, afrag[1][kb],
                                                       false, bfrag,
                                                       (short)0, d1, false, false);
        }
        __syncthreads();
        // D -> LDS (next B operand) and -> global z_{k+1}.
#pragma unroll
        for (int j = 0; j < 8; ++j) {
            int m = j + hi * 8;
            zt[m * 16 + lo] = d0[j];
            zt[(16 + m) * 16 + lo] = d1[j];
            long zb = (long)(b0 + lo) * (NCHUNK * 32) + (k + 1) * 32;
            zbuf[zb + m] = d0[j];
            zbuf[zb + 16 + m] = d1[j];
        }
        __syncthreads();
    }
}

// ---------------------------------------------------------------------------
// Kernel 5: pass3 — exact chunk re-scan from z_start + fused x->out copy.
// One lane per (b, chunk); writes all 11 output channels for t in [24, T).
// ---------------------------------------------------------------------------
__global__ __launch_bounds__(SBLOCK) void pass3_kernel(const float* __restrict__ x,
                                                       const float* __restrict__ alpha,
                                                       const float* __restrict__ gamma,
                                                       const float* __restrict__ zbuf,
                                                       float* __restrict__ out) {
    __shared__ float sbuf[SEAS * SBLOCK];
    const int tid = threadIdx.x;
    const int gid = blockIdx.x * SBLOCK + tid;
    const int b = gid >> 4;
    const int k = gid & 15;
    if (b >= BATCH) return;
    const float a = alpha[0], g = gamma[0];
    const float oma = 1.0f - a, omg = 1.0f - g;
    const int t0 = SEAS + k * CHUNK;
    const int tend = (t0 + CHUNK < TLEN) ? (t0 + CHUNK) : TLEN;
    const float* z = zbuf + ((long)b * NCHUNK + k) * 32;
    float l = z[0];
#pragma unroll
    for (int i = 0; i < SEAS; ++i)               // s_{t0-24+i} -> slot (t0+i)%24
        sbuf[((t0 + i) % SEAS) * SBLOCK + tid] = z[1 + i];
    int slot = t0 % SEAS;
    const long base = (long)b * TLEN;
    for (int t = t0; t < tend; ++t) {
        long bt = base + t;
        float4 xa = *(const float4*)(x + bt * 8);
        float4 xb = *(const float4*)(x + bt * 8 + 4);
        float p = xa.x;
        float sp = sbuf[slot * SBLOCK + tid];
        l = __builtin_fmaf(oma, l, a * (p - sp));
        float sv = __builtin_fmaf(omg, sp, g * (p - l));
        float y = p - sv;
        sbuf[slot * SBLOCK + tid] = sv;
        float* o = out + bt * 11;
        o[0] = y;
        o[1] = xa.x; o[2] = xa.y; o[3] = xa.z; o[4] = xa.w;
        o[5] = xb.x; o[6] = xb.y; o[7] = xb.z; o[8] = xb.w;
        o[9] = l; o[10] = sv;
        slot = (slot + 1 == SEAS) ? 0 : slot + 1;
    }
}

// ---------------------------------------------------------------------------
extern "C" void kernel_launch(void* const* d_in, const int* in_sizes, int n_in,
                              void* d_out, int out_size, void* d_ws, size_t ws_size,
                              hipStream_t stream) {
    const float* x     = (const float*)d_in[0];   // (512, 8192, 8) f32
    const float* alpha = (const float*)d_in[1];   // (1,) f32
    const float* gamma = (const float*)d_in[2];   // (1,) f32
    // d_in[3] = season_length (int, ==24); structurally hard-coded.
    float* out = (float*)d_out;                   // (512, 8192, 11) f32
    float* ws  = (float*)d_ws;

    float* wsA   = ws;                              // 1024 floats: M^512 (32x32)
    float* wsLvl = ws + 1024;                       // 512 floats : init levels
    float* wsC   = ws + 2048;                       // 512*16*32  : chunk c vectors
    float* wsZ   = wsC + (long)BATCH * NCHUNK * 32; // 512*16*32  : chunk start states

    matpow_kernel<<<1, 256, 0, stream>>>(alpha, gamma, wsA);
    head_kernel<<<BATCH, 32, 0, stream>>>(x, out, wsLvl);
    pass1_kernel<<<(BATCH * NCHUNK) / SBLOCK, SBLOCK, 0, stream>>>(x, alpha, gamma, wsC);
    combine_kernel<<<BATCH / 16, 32, 0, stream>>>(wsA, wsLvl, wsC, wsZ);
    pass3_kernel<<<(BATCH * NCHUNK) / SBLOCK, SBLOCK, 0, stream>>>(x, alpha, gamma, wsZ, out);
}